// GATEncoder_28209345200424
// MI455X (gfx1250) — compile-verified
//
#include <hip/hip_runtime.h>

typedef float v2f __attribute__((ext_vector_type(2)));
typedef float v8f __attribute__((ext_vector_type(8)));

#define N_NODES 50000          // == 16 * 3125: no partial 16-row tiles
#define FDIM    128
#define N_EDGES 800000
#define E_TOT   (N_EDGES + N_NODES)
#define WT_STRIDE (FDIM + 1)   // padded LDS stride to avoid bank conflicts

// ---------------- helpers ----------------

__device__ __forceinline__ unsigned f2ord(float f) {
    unsigned u = __float_as_uint(f);
    return (u >> 31) ? ~u : (u | 0x80000000u);   // monotone: f < g  <=>  f2ord(f) < f2ord(g)
}
__device__ __forceinline__ float ord2f(unsigned u) {
    return (u & 0x80000000u) ? __uint_as_float(u ^ 0x80000000u) : __uint_as_float(~u);
}
__device__ __forceinline__ void edge_nodes(const int* __restrict__ ei, int e, int& s, int& d) {
    if (e < N_EDGES) { s = ei[e]; d = ei[N_EDGES + e]; }
    else             { s = d = e - N_EDGES; }       // appended self-loops
}

// ---------------- kernels ----------------

__global__ void zero_u32(unsigned* __restrict__ p, int n) {
    int i = blockIdx.x * blockDim.x + threadIdx.x;
    if (i < n) p[i] = 0u;
}

// h[N,128] = x[N,128] @ W[128,128] using V_WMMA_F32_16X16X4_F32.
// Block = 256 threads = 8 waves; wave w computes rows [blk*128 + w*16, +16) x all 128 cols.
// W^T staged in LDS (padded stride) so each B-fragment is one paired ds load.
// N_NODES is a multiple of 16, so no per-row guards are needed anywhere.
__global__ void gemm_wmma(const float* __restrict__ x, const float* __restrict__ W,
                          float* __restrict__ h) {
    extern __shared__ float wt[];                 // [128][129]  W^T: wt[n*129 + k] = W[k*128 + n]
    const int tid = threadIdx.x;
    for (int i = tid; i < FDIM * FDIM; i += 256) {
        int k = i >> 7, n = i & 127;
        wt[n * WT_STRIDE + k] = W[i];
    }
    __syncthreads();

    const int wave = tid >> 5;
    const int lane = tid & 31;
    const int half = lane >> 4;                   // 0: lanes 0-15 (K=0,1)  1: lanes 16-31 (K=2,3)
    const int l16  = lane & 15;
    const int rowBase = (blockIdx.x * 8 + wave) * 16;
    if (rowBase >= N_NODES) return;               // only whole tiles exist below this

    // A fragment source: this lane supplies row (rowBase + l16), K-pair at half*2
    const float* xrow = x + (size_t)(rowBase + l16) * FDIM + half * 2;
    // B fragment source in LDS
    const float* wb0 = wt + half * 2;

    v8f acc[8];
    #pragma unroll
    for (int t = 0; t < 8; ++t)
        #pragma unroll
        for (int j = 0; j < 8; ++j) acc[t][j] = 0.0f;

    for (int ks = 0; ks < 32; ++ks) {             // K = 128 = 32 * 4
        v2f a = *(const v2f*)(xrow + ks * 4);     // one 8B global load (always 8B aligned)
        const float* wb = wb0 + ks * 4;
        #pragma unroll
        for (int t = 0; t < 8; ++t) {             // 8 column tiles of 16
            const float* bp = wb + (t * 16 + l16) * WT_STRIDE;
            v2f b;                                 // pairs into ds_load_2addr_b32
            b.x = bp[0];
            b.y = bp[1];
            acc[t] = __builtin_amdgcn_wmma_f32_16x16x4_f32(
                false, a, false, b, (short)0, acc[t], false, false);
        }
    }

    // 16x16 FP32 C/D layout: VGPR g, lane -> row = g + 8*half, col = l16
    float* hout = h + (size_t)(rowBase + half * 8) * FDIM + l16;
    #pragma unroll
    for (int t = 0; t < 8; ++t) {
        #pragma unroll
        for (int g = 0; g < 8; ++g) {
            hout[g * FDIM + t * 16] = acc[t][g];  // unconditional coalesced stores
        }
    }
}

// alpha_s[n,h] = <h[n, h*C : h*C+C], a_src[h]>, same for alpha_d. One thread per (n, head).
__global__ void alpha_kernel(const float* __restrict__ hf,
                             const float* __restrict__ a_s, const float* __restrict__ a_d,
                             float* __restrict__ as_out, float* __restrict__ ad_out,
                             int H, int C) {
    int i = blockIdx.x * blockDim.x + threadIdx.x;
    if (i >= N_NODES * H) return;
    int n = i / H, hh = i % H;
    const float* hp  = hf + n * FDIM + hh * C;
    const float* asp = a_s + hh * C;
    const float* adp = a_d + hh * C;
    float ss = 0.f, sd = 0.f;
    for (int c = 0; c < C; c += 4) {
        float4 hv = *(const float4*)(hp + c);
        float4 av = *(const float4*)(asp + c);
        float4 dv = *(const float4*)(adp + c);
        ss += hv.x * av.x + hv.y * av.y + hv.z * av.z + hv.w * av.w;
        sd += hv.x * dv.x + hv.y * dv.y + hv.z * dv.z + hv.w * dv.w;
    }
    as_out[i] = ss;
    ad_out[i] = sd;
}

__global__ void edge_max_kernel(const int* __restrict__ ei,
                                const float* __restrict__ as, const float* __restrict__ ad,
                                unsigned* __restrict__ emax_u, int H) {
    int e = blockIdx.x * blockDim.x + threadIdx.x;
    if (e >= E_TOT) return;
    int s, d; edge_nodes(ei, e, s, d);
    for (int hh = 0; hh < H; ++hh) {
        float el = as[s * H + hh] + ad[d * H + hh];
        el = el > 0.f ? el : 0.2f * el;                 // leaky_relu
        atomicMax(&emax_u[d * H + hh], f2ord(el));      // -> GLOBAL_ATOMIC_MAX_U32
    }
}

__global__ void edge_sum_kernel(const int* __restrict__ ei,
                                const float* __restrict__ as, const float* __restrict__ ad,
                                const unsigned* __restrict__ emax_u,
                                float* __restrict__ denom, int H) {
    int e = blockIdx.x * blockDim.x + threadIdx.x;
    if (e >= E_TOT) return;
    int s, d; edge_nodes(ei, e, s, d);
    for (int hh = 0; hh < H; ++hh) {
        float el = as[s * H + hh] + ad[d * H + hh];
        el = el > 0.f ? el : 0.2f * el;
        float p = __expf(el - ord2f(emax_u[d * H + hh]));
        atomicAdd(&denom[d * H + hh], p);               // -> GLOBAL_ATOMIC_ADD_F32
    }
}

// One wave (32 lanes) per edge: lane owns 4 consecutive channels (float4 gather + 4 f32 atomics).
__global__ void aggregate_kernel(const int* __restrict__ ei, const float* __restrict__ hf,
                                 const float* __restrict__ as, const float* __restrict__ ad,
                                 const unsigned* __restrict__ emax_u,
                                 const float* __restrict__ denom,
                                 float* __restrict__ acc, int H, int C) {
    int t = blockIdx.x * blockDim.x + threadIdx.x;
    int e = t >> 5;
    if (e >= E_TOT) return;
    int chunk = t & 31;
    int c0 = chunk * 4;
    int s, d; edge_nodes(ei, e, s, d);
    int hh = c0 / C;                                    // head owning this channel chunk
    float el = as[s * H + hh] + ad[d * H + hh];
    el = el > 0.f ? el : 0.2f * el;
    float alpha = __expf(el - ord2f(emax_u[d * H + hh])) / denom[d * H + hh];
    float4 hv = *(const float4*)(hf + s * FDIM + c0);
    float* op = acc + d * FDIM + c0;
    atomicAdd(op + 0, hv.x * alpha);
    atomicAdd(op + 1, hv.y * alpha);
    atomicAdd(op + 2, hv.z * alpha);
    atomicAdd(op + 3, hv.w * alpha);
}

__global__ void bias_relu_kernel(float* __restrict__ acc, const float* __restrict__ b) {
    int i = blockIdx.x * blockDim.x + threadIdx.x;
    if (i >= N_NODES * FDIM) return;
    float v = acc[i] + b[i & 127];
    acc[i] = v > 0.f ? v : 0.f;
}

// ---------------- launch ----------------

extern "C" void kernel_launch(void* const* d_in, const int* in_sizes, int n_in,
                              void* d_out, int out_size, void* d_ws, size_t ws_size,
                              hipStream_t stream) {
    const float* x  = (const float*)d_in[0];
    const int*   ei = (const int*)d_in[1];
    const float* Ws[3] = {(const float*)d_in[2], (const float*)d_in[6],  (const float*)d_in[10]};
    const float* As[3] = {(const float*)d_in[3], (const float*)d_in[7],  (const float*)d_in[11]};
    const float* Ad[3] = {(const float*)d_in[4], (const float*)d_in[8],  (const float*)d_in[12]};
    const float* Bs[3] = {(const float*)d_in[5], (const float*)d_in[9],  (const float*)d_in[13]};
    const int Hs[3] = {4, 4, 1};

    float* ws      = (float*)d_ws;
    float* hfeat   = ws;                               // N*128
    float* buf1    = ws + (size_t)N_NODES * FDIM;      // N*128  (layer0 out / layer1 in)
    float* buf2    = ws + (size_t)2 * N_NODES * FDIM;  // N*128  (layer1 out / layer2 in)
    float* alpha_s = ws + (size_t)3 * N_NODES * FDIM;  // N*4
    float* alpha_d = alpha_s + (size_t)N_NODES * 4;    // N*4
    unsigned* emax_u = (unsigned*)(alpha_d + (size_t)N_NODES * 4);   // N*4
    float* denom   = (float*)(emax_u + (size_t)N_NODES * 4);         // N*4

    const int row_tiles   = (N_NODES + 15) / 16;       // 3125
    const int gemm_blocks = (row_tiles + 7) / 8;       // 391
    const int nf          = N_NODES * FDIM;            // 6.4M
    const int stat_n      = N_NODES * 4 * 2;           // emax_u + denom (contiguous)

    const float* cur = x;
    float* outs[3] = {buf1, buf2, (float*)d_out};

    for (int L = 0; L < 3; ++L) {
        const int H = Hs[L], C = FDIM / H;
        float* acc = outs[L];

        zero_u32<<<(nf + 255) / 256, 256, 0, stream>>>((unsigned*)acc, nf);
        zero_u32<<<(stat_n + 255) / 256, 256, 0, stream>>>(emax_u, stat_n);

        gemm_wmma<<<gemm_blocks, 256, FDIM * WT_STRIDE * sizeof(float), stream>>>(cur, Ws[L], hfeat);

        int na = N_NODES * H;
        alpha_kernel<<<(na + 255) / 256, 256, 0, stream>>>(hfeat, As[L], Ad[L],
                                                           alpha_s, alpha_d, H, C);

        edge_max_kernel<<<(E_TOT + 255) / 256, 256, 0, stream>>>(ei, alpha_s, alpha_d, emax_u, H);
        edge_sum_kernel<<<(E_TOT + 255) / 256, 256, 0, stream>>>(ei, alpha_s, alpha_d,
                                                                 emax_u, denom, H);

        int agg_threads = E_TOT * 32;                   // 27.2M
        aggregate_kernel<<<(agg_threads + 255) / 256, 256, 0, stream>>>(
            ei, hfeat, alpha_s, alpha_d, emax_u, denom, acc, H, C);

        bias_relu_kernel<<<(nf + 255) / 256, 256, 0, stream>>>(acc, Bs[L]);

        cur = acc;
    }
}